// Inference_and_Generation_8418135900732
// MI455X (gfx1250) — compile-verified
//
#include <hip/hip_runtime.h>
#include <math.h>

// ---- problem constants (from reference) ----
#define B_SZ   8
#define DZ     32
#define NW     48
#define NH     48
#define NBOX   (NW*NH)        // 2304 boxes per batch
#define KMAX   25
#define PIX    (384*384)      // 147456 pixels per (k,b) plane
#define SCORE_T 0.1f
#define OVER_T  0.3f

typedef __attribute__((ext_vector_type(2))) float v2f;
typedef __attribute__((ext_vector_type(8))) float v8f;

__device__ __forceinline__ float sigmoidf(float x) { return 1.0f / (1.0f + expf(-x)); }

// ============================================================================
// Kernel A: fused 1x1-conv GEMM via V_WMMA_F32_16X16X4_F32 + box decode.
//
// Logical GEMM: D[16 x 16-tile] = A[16 x 64] * B[64 x 16-tile]
//   A rows 0..3  = [ w_zwhere[o][0..31] | 0 ]      (K 0..31  -> zwhere path)
//   A row  4     = [ 0 | w_logit[0..31] ]          (K 32..63 -> logit path)
//   A rows 5..15 = 0 (padding)
//   B rows 0..31  = zwhere_sample[b, d, n]   (n = 16-wide column tile)
//   B rows 32..63 = logit_sample [b, d, n]
// K split into 16 steps of 4 for the f32 WMMA.
//
// The padded 16x64 weight matrix (4 KB) is built once in LDS per block so the
// WMMA loop has no divergent global loads: ds_load_b64 for A, clause-paired
// global_load_b32 for B, one v_wmma per step.
//
// VGPR layouts per CDNA5 ISA 7.12.2:
//   A 16x4: lane L holds row M=L%16; lanes<16 -> K={k0,k0+1}, lanes>=16 -> K={k0+2,k0+3}
//   B 4x16: lane L holds col N=L%16; same K striping as A
//   C/D 16x16: lanes<16 -> N=lane, VGPR r = row M=r (rows 0..4 are what we need)
// ============================================================================
__global__ __launch_bounds__(128)
void conv_box_kernel(const float* __restrict__ zs, const float* __restrict__ wz,
                     const float* __restrict__ bz, const float* __restrict__ ls,
                     const float* __restrict__ wl, const float* __restrict__ bl,
                     float* __restrict__ prob, float* __restrict__ x1o,
                     float* __restrict__ y1o, float* __restrict__ x2o,
                     float* __restrict__ y2o)
{
    __shared__ float Wlds[16 * 64];      // padded A matrix, row-major 16x64

    // cooperative fill: 128 threads x 8 entries
    for (int i = threadIdx.x; i < 16 * 64; i += 128) {
        const int r = i >> 6;            // row 0..15
        const int c = i & 63;            // col 0..63
        float v = 0.0f;
        if (r < 4) { if (c < DZ) v = wz[r * DZ + c]; }
        else if (r == 4) { if (c >= DZ) v = wl[c - DZ]; }
        Wlds[i] = v;
    }
    __syncthreads();

    const int lane    = threadIdx.x & 31;
    const int wave    = blockIdx.x * (blockDim.x >> 5) + (threadIdx.x >> 5);
    const int b       = wave / (NBOX / 16);
    const int tile    = wave % (NBOX / 16);
    const int nbase   = tile * 16;
    const int halfsel = lane >> 4;      // 0: lanes 0-15, 1: lanes 16-31
    const int mrow    = lane & 15;      // A: row M   /   B: column N
    const int nn      = nbase + mrow;   // global box index this lane loads for B

    v8f acc = {};                        // C starts at 0

#pragma unroll
    for (int kk = 0; kk < 16; ++kk) {
        const int k = kk * 4 + halfsel * 2;   // this lane holds K = {k, k+1}

        // ---- A operand from LDS (8B aligned pair, no divergence) ----
        const v2f av = *(const v2f*)&Wlds[mrow * 64 + k];

        // ---- B operand (activations, K-stacked zwhere|logit) ----
        const float* xp = (k < DZ)
            ? (zs + ((size_t)b * DZ + k)        * NBOX)
            : (ls + ((size_t)b * DZ + (k - DZ)) * NBOX);
        v2f bv = {xp[nn], xp[NBOX + nn]};

        acc = __builtin_amdgcn_wmma_f32_16x16x4_f32(false, av, false, bv,
                                                    (short)0, acc, false, false);
    }

    // rows 0..4 of D live in lanes 0..15, VGPRs 0..4 (col N = lane)
    if (lane < 16) {
        const int n  = nbase + lane;
        const float tx = sigmoidf(acc[0] + bz[0]);
        const float ty = sigmoidf(acc[1] + bz[1]);
        const float tw = sigmoidf(acc[2] + bz[2]);
        const float th = sigmoidf(acc[3] + bz[3]);
        const float pr = sigmoidf(acc[4] + bl[0]);
        const float ix = (float)(n / NH);
        const float iy = (float)(n % NH);
        const float bx = 8.0f * (ix + tx);          // 384/48
        const float by = 8.0f * (iy + ty);
        const float bw = 12.0f + 36.0f * tw;        // SIZE_MIN + range*t
        const float bh = 12.0f + 36.0f * th;
        const int o = b * NBOX + n;
        prob[o] = pr;
        x1o[o] = bx - 0.5f * bw;  x2o[o] = bx + 0.5f * bw;
        y1o[o] = by - 0.5f * bh;  y2o[o] = by + 0.5f * bh;
    }
}

// ============================================================================
// Kernel B: greedy NMS, one workgroup per batch. Boxes live in LDS.
// Produces prob_few[k*B+b] = probT[b, sel_k] * mask[b, sel_k].
// ============================================================================
__global__ __launch_bounds__(256)
void nms_kernel(const float* __restrict__ prob, const float* __restrict__ x1f,
                const float* __restrict__ y1f, const float* __restrict__ x2f,
                const float* __restrict__ y2f, float* __restrict__ prob_few)
{
    __shared__ float sp[NBOX], sx1[NBOX], sy1[NBOX], sx2[NBOX], sy2[NBOX];
    __shared__ float smask[NBOX];   // masked prob; -inf == not "possible"
    __shared__ float mk[NBOX];
    __shared__ float rval[256];
    __shared__ int   ridx[256];
    __shared__ int   s_sel;
    __shared__ int   s_idx[KMAX];
    __shared__ int   s_ok[KMAX];

    const int b = blockIdx.x;
    const int t = threadIdx.x;

    for (int n = t; n < NBOX; n += 256) {
        const float p = prob[b * NBOX + n];
        sp[n] = p;
        sx1[n] = x1f[b * NBOX + n];  sy1[n] = y1f[b * NBOX + n];
        sx2[n] = x2f[b * NBOX + n];  sy2[n] = y2f[b * NBOX + n];
        smask[n] = (p > SCORE_T) ? p : -INFINITY;
        mk[n] = 0.0f;
    }
    __syncthreads();

    for (int k = 0; k < KMAX; ++k) {
        // --- argmax over masked prob, first index wins ties (jnp.argmax) ---
        float bv = -INFINITY; int bi = 0;
        for (int n = t; n < NBOX; n += 256) {
            const float v = smask[n];
            if (v > bv) { bv = v; bi = n; }
        }
        rval[t] = bv; ridx[t] = bi;
        __syncthreads();
        for (int s = 128; s > 0; s >>= 1) {
            if (t < s) {
                const float v2 = rval[t + s]; const int i2 = ridx[t + s];
                if (v2 > rval[t] || (v2 == rval[t] && i2 < ridx[t])) {
                    rval[t] = v2; ridx[t] = i2;
                }
            }
            __syncthreads();
        }
        if (t == 0) {
            const int sel = ridx[0];
            s_sel = sel;
            s_idx[k] = sel;
            s_ok[k] = (smask[sel] != -INFINITY) ? 1 : 0;   // possible[sel]
        }
        __syncthreads();

        // --- suppress everything with IoU(sel, n) > OVER_T (incl. self) ---
        const int sel = s_sel;
        const float ax1 = sx1[sel], ay1 = sy1[sel], ax2 = sx2[sel], ay2 = sy2[sel];
        const float aarea = (ax2 - ax1) * (ay2 - ay1);
        for (int n = t; n < NBOX; n += 256) {
            const float xi1 = fmaxf(ax1, sx1[n]);
            const float yi1 = fmaxf(ay1, sy1[n]);
            const float xi2 = fminf(ax2, sx2[n]);
            const float yi2 = fminf(ay2, sy2[n]);
            const float inter = fmaxf(xi2 - xi1, 0.0f) * fmaxf(yi2 - yi1, 0.0f);
            const float narea = (sx2[n] - sx1[n]) * (sy2[n] - sy1[n]);
            const float iou = inter / fmaxf(aarea + narea - inter, 1e-8f);
            if (iou > OVER_T) smask[n] = -INFINITY;
        }
        __syncthreads();
    }

    // mask[n] = 1 where a valid selection landed; then gather prob_few
    if (t < KMAX) { if (s_ok[t]) mk[s_idx[t]] = 1.0f; }
    __syncthreads();
    if (t < KMAX) {
        const int idx = s_idx[t];
        prob_few[t * B_SZ + b] = sp[idx] * mk[idx];
    }
}

// ============================================================================
// Kernel C: bandwidth-bound finale. One thread per float4 of pixels.
// Reads weight_logits exactly once (25 softplus values kept in registers),
// writes output once: ~236 MB total traffic -> ~10 us at 23.3 TB/s.
// ============================================================================
__global__ __launch_bounds__(256)
void mask_kernel(const float* __restrict__ wlg, const float* __restrict__ prob_few,
                 float* __restrict__ out)
{
    const int tid = blockIdx.x * blockDim.x + threadIdx.x;
    const int P4  = PIX / 4;                 // 36864 float4 per plane
    const int b   = tid / P4;
    const int p4  = tid % P4;
    const size_t base = (size_t)b * PIX + (size_t)p4 * 4;

    float4 sp[KMAX];
    float4 s = make_float4(0.0f, 0.0f, 0.0f, 0.0f);

#pragma unroll
    for (int k = 0; k < KMAX; ++k) {
        const float4 w = *(const float4*)(wlg + (size_t)k * B_SZ * PIX + base);
        float4 v;   // numerically stable softplus: max(x,0) + log1p(exp(-|x|))
        v.x = fmaxf(w.x, 0.0f) + log1pf(expf(-fabsf(w.x)));
        v.y = fmaxf(w.y, 0.0f) + log1pf(expf(-fabsf(w.y)));
        v.z = fmaxf(w.z, 0.0f) + log1pf(expf(-fabsf(w.z)));
        v.w = fmaxf(w.w, 0.0f) + log1pf(expf(-fabsf(w.w)));
        sp[k] = v;
        s.x += v.x; s.y += v.y; s.z += v.z; s.w += v.w;
    }

    float4 scale;   // tanh(sum) / max(sum, 1e-6)
    scale.x = tanhf(s.x) / fmaxf(s.x, 1e-6f);
    scale.y = tanhf(s.y) / fmaxf(s.y, 1e-6f);
    scale.z = tanhf(s.z) / fmaxf(s.z, 1e-6f);
    scale.w = tanhf(s.w) / fmaxf(s.w, 1e-6f);

#pragma unroll
    for (int k = 0; k < KMAX; ++k) {
        const float pf = prob_few[k * B_SZ + b];
        float4 o;
        o.x = pf * scale.x * sp[k].x;
        o.y = pf * scale.y * sp[k].y;
        o.z = pf * scale.z * sp[k].z;
        o.w = pf * scale.w * sp[k].w;
        *(float4*)(out + (size_t)k * B_SZ * PIX + base) = o;
    }
}

// ============================================================================
extern "C" void kernel_launch(void* const* d_in, const int* in_sizes, int n_in,
                              void* d_out, int out_size, void* d_ws, size_t ws_size,
                              hipStream_t stream)
{
    const float* zs  = (const float*)d_in[0];   // zwhere_sample (B,DZ,48,48)
    const float* wz  = (const float*)d_in[1];   // w_zwhere (4,DZ)
    const float* bz  = (const float*)d_in[2];   // b_zwhere (4,)
    const float* ls  = (const float*)d_in[3];   // logit_sample (B,DZ,48,48)
    const float* wl  = (const float*)d_in[4];   // w_logit (1,DZ)
    const float* bl  = (const float*)d_in[5];   // b_logit (1,)
    const float* wlg = (const float*)d_in[6];   // weight_logits (K,B,1,384,384)
    float* out = (float*)d_out;

    float* prob     = (float*)d_ws;             // B*NBOX
    float* x1       = prob + B_SZ * NBOX;
    float* y1       = x1   + B_SZ * NBOX;
    float* x2       = y1   + B_SZ * NBOX;
    float* y2       = x2   + B_SZ * NBOX;
    float* prob_few = y2   + B_SZ * NBOX;       // KMAX*B

    // A: 1152 waves (one per 16-box tile per batch), 4 waves/block
    conv_box_kernel<<<(B_SZ * (NBOX / 16)) / 4, 128, 0, stream>>>(
        zs, wz, bz, ls, wl, bl, prob, x1, y1, x2, y2);

    // B: one workgroup per batch
    nms_kernel<<<B_SZ, 256, 0, stream>>>(prob, x1, y1, x2, y2, prob_few);

    // C: one thread per float4 of output pixels
    mask_kernel<<<(B_SZ * (PIX / 4)) / 256, 256, 0, stream>>>(wlg, prob_few, out);
}